// BuseE_781684048755
// MI455X (gfx1250) — compile-verified
//
#include <hip/hip_runtime.h>
#include <math.h>

typedef __attribute__((ext_vector_type(2))) float v2f;
typedef __attribute__((ext_vector_type(8))) float v8f;

#define MIN_NORM 1e-15f
#define MARGIN   9.0f
#define DIM      64
#define Bq       1024
#define NC       1024
#define ST       68   // padded LDS row stride (floats) to break bank conflicts

// ---------------- wave32 helpers ----------------
__device__ __forceinline__ float wave_sum(float v) {
#pragma unroll
    for (int off = 16; off > 0; off >>= 1) v += __shfl_xor(v, off, 32);
    return v;
}

__device__ __forceinline__ float2 mobius_add_pair(float2 x, float2 y) {
    float x2 = wave_sum(x.x * x.x + x.y * x.y);
    float y2 = wave_sum(y.x * y.x + y.y * y.y);
    float xy = wave_sum(x.x * y.x + x.y * y.y);
    float c1  = 1.0f + 2.0f * xy + y2;
    float c2  = 1.0f - x2;
    float den = 1.0f + 2.0f * xy + x2 * y2;
    float inv = 1.0f / fmaxf(den, MIN_NORM);
    float2 r;
    r.x = (c1 * x.x + c2 * y.x) * inv;
    r.y = (c1 * x.y + c2 * y.y) * inv;
    return r;
}

__device__ __forceinline__ float2 expmap0_pair(float2 u, float sq_full) {
    float un = fmaxf(sqrtf(sq_full), MIN_NORM);
    float s  = tanhf(un) / un;
    float2 r; r.x = u.x * s; r.y = u.y * s;
    return r;
}

// ============ Kernel 1: per-query head transform (one wave32 per b) ============
__global__ __launch_bounds__(32) void head_kernel(
    const int* __restrict__ u_idx, const int* __restrict__ r_idx,
    const float* __restrict__ emb, const float* __restrict__ rel_diag,
    const float* __restrict__ rb1, const float* __restrict__ rb2,
    const float* __restrict__ bias_head, const float* __restrict__ sigma,
    float* __restrict__ ws_head, float4* __restrict__ ws_scal)
{
    const int b    = blockIdx.x;
    const int lane = threadIdx.x;          // 0..31, owns dims 2*lane, 2*lane+1
    const int u = u_idx[b];
    const int r = r_idx[b];

    float2 e  = *(const float2*)(emb      + (size_t)u * DIM + 2 * lane);
    float2 y1 = *(const float2*)(rb1      + (size_t)r * DIM + 2 * lane);
    float2 y2 = *(const float2*)(rb2      + (size_t)r * DIM + 2 * lane);
    float2 g  = *(const float2*)(rel_diag + (size_t)r * DIM + 2 * lane);

    float ne = wave_sum(e.x * e.x + e.y * e.y);
    float n1 = wave_sum(y1.x * y1.x + y1.y * y1.y);
    float n2 = wave_sum(y2.x * y2.x + y2.y * y2.y);

    float2 h  = expmap0_pair(e,  ne);
    float2 b1 = expmap0_pair(y1, n1);
    float2 b2 = expmap0_pair(y2, n2);

    h = mobius_add_pair(h, b1);

    // Givens rotation on (2*lane, 2*lane+1) pair
    float gn = fmaxf(sqrtf(g.x * g.x + g.y * g.y), MIN_NORM);
    float gx = g.x / gn, gy = g.y / gn;
    float2 hr;
    hr.x = gx * h.x - gy * h.y;
    hr.y = gy * h.x + gx * h.y;

    h = mobius_add_pair(hr, b2);

    float hsq = wave_sum(h.x * h.x + h.y * h.y);

    *(float2*)(ws_head + (size_t)b * DIM + 2 * lane) = h;
    if (lane == 0) {
        float log_den_h = logf(fmaxf(1.0f - hsq, MIN_NORM));
        float sg = 1.0f / (1.0f + expf(-sigma[r]));
        float4 sc;
        sc.x = hsq; sc.y = log_den_h; sc.z = sg; sc.w = bias_head[u];
        ws_scal[b] = sc;
    }
}

// Issue 8 async b128 gathers: one half entity row (128 B) straight into LDS.
// INST_OFFSET is added to BOTH the global and LDS address (ISA 15.18.3), and
// each 128 B half-row is contiguous in both spaces, so plain offsets work.
__device__ __forceinline__ void issue_async_halfrow(unsigned lds_off, const float* gsrc) {
    asm volatile(
        "global_load_async_to_lds_b128 %0, %1, off\n\t"
        "global_load_async_to_lds_b128 %0, %1, off offset:16\n\t"
        "global_load_async_to_lds_b128 %0, %1, off offset:32\n\t"
        "global_load_async_to_lds_b128 %0, %1, off offset:48\n\t"
        "global_load_async_to_lds_b128 %0, %1, off offset:64\n\t"
        "global_load_async_to_lds_b128 %0, %1, off offset:80\n\t"
        "global_load_async_to_lds_b128 %0, %1, off offset:96\n\t"
        "global_load_async_to_lds_b128 %0, %1, off offset:112"
        :: "v"(lds_off), "v"(gsrc) : "memory");
}

// ============ Kernel 2: scoring with f32 WMMA dot products ============
// One block per b (256 threads = 8 waves). Each wave handles 8 tiles of 16
// tails, double-buffered via async direct-to-LDS gathers (ASYNCcnt).
__global__ __launch_bounds__(256) void score_kernel(
    const int* __restrict__ v_idx, const float* __restrict__ emb,
    const float* __restrict__ bias_tail,
    const float* __restrict__ ws_head, const float4* __restrict__ ws_scal,
    float* __restrict__ out)
{
    __shared__ float s_head[DIM];
    __shared__ __align__(16) float s_tail[2][8][16 * ST];
    __shared__ float s_dot[8][16];

    const int b    = blockIdx.x;
    const int tid  = threadIdx.x;
    const int wave = tid >> 5;
    const int lane = tid & 31;

    if (tid < DIM) s_head[tid] = ws_head[(size_t)b * DIM + tid];
    const float4 sc = ws_scal[b];   // {hsq, log_den_h, sig, bias_head}
    __syncthreads();

    const int m    = lane >> 1;            // tail row this lane gathers
    const int half = lane & 1;             // which 32-float half
    const int am   = lane & 15;            // A-matrix row M this lane feeds
    const int koff = (lane >> 4) * 2;      // lanes 16-31 feed K+2,K+3

    // prologue: gather tile 0 into buffer 0
    {
        const int v = v_idx[(size_t)b * NC + (wave * 8) * 16 + m];
        issue_async_halfrow(
            (unsigned)(uintptr_t)&s_tail[0][wave][m * ST + half * 32],
            emb + (size_t)v * DIM + half * 32);
    }

    for (int it = 0; it < 8; ++it) {
        const int cur   = it & 1;
        const int cbase = (wave * 8 + it) * 16;

        if (it < 7) {   // gather next tile into the other buffer
            const int v = v_idx[(size_t)b * NC + cbase + 16 + m];
            issue_async_halfrow(
                (unsigned)(uintptr_t)&s_tail[cur ^ 1][wave][m * ST + half * 32],
                emb + (size_t)v * DIM + half * 32);
            asm volatile("s_wait_asynccnt 0x8" ::: "memory"); // tile `it` done (in-order)
        } else {
            asm volatile("s_wait_asynccnt 0x0" ::: "memory");
        }

        // ---- 16 dots via V_WMMA_F32_16X16X4_F32, K=64 in 16 steps ----
        // A: 16 tails x 4 (lane L<16 -> M=L, K={k,k+1}; L>=16 -> M=L-16, K={k+2,k+3})
        // B: head chunk broadcast across all 16 columns (same half-lane K split)
        const float* arow = &s_tail[cur][wave][am * ST + koff];
        v8f acc = {};
#pragma unroll
        for (int k0 = 0; k0 < DIM; k0 += 4) {
            v2f a, bb;
            a.x  = arow[k0];
            a.y  = arow[k0 + 1];
            bb.x = s_head[k0 + koff];
            bb.y = s_head[k0 + koff + 1];
            acc = __builtin_amdgcn_wmma_f32_16x16x4_f32(
                false, a, false, bb, (short)0, acc, false, false);
        }

        // All D-columns identical: lane 0 holds dots M=0..7, lane 16 holds M=8..15
        if (lane == 0 || lane == 16) {
            const int mb = (lane >> 4) * 8;
#pragma unroll
            for (int i = 0; i < 8; ++i) s_dot[wave][mb + i] = acc[i];
        }
        asm volatile("s_wait_dscnt 0" ::: "memory");

        // ---- epilogue: lanes 0..15 finish one tail each ----
        if (lane < 16) {
            const float* row = &s_tail[cur][wave][lane * ST];
            float n2raw = 0.0f;
#pragma unroll
            for (int k = 0; k < DIM; k += 4) {
                float4 q = *(const float4*)(row + k);
                n2raw += q.x * q.x + q.y * q.y + q.z * q.z + q.w * q.w;
            }
            const float draw = s_dot[wave][lane];

            // deferred expmap0 on tail: tail = (tanh(n)/n) * raw
            const float un    = fmaxf(sqrtf(n2raw), MIN_NORM);
            const float th    = tanhf(un);
            const float scale = th / un;
            const float tsq   = th * th;             // ||tail||^2
            const float dsc   = scale * draw;        // <head, tail>

            const float num   = fmaxf(sc.x + tsq - 2.0f * dsc, MIN_NORM);
            const float den_t = fmaxf(1.0f - tsq, MIN_NORM);

            // dist = sig*log(num/den_t) + (1-sig)*log(num/den_h)
            const float dist = logf(num) - sc.z * logf(den_t) - (1.0f - sc.z) * sc.y;

            const int c  = cbase + lane;
            const int vv = v_idx[(size_t)b * NC + c];
            out[(size_t)b * NC + c] = MARGIN - dist + sc.w + bias_tail[vv];
        }
    }
}

extern "C" void kernel_launch(void* const* d_in, const int* in_sizes, int n_in,
                              void* d_out, int out_size, void* d_ws, size_t ws_size,
                              hipStream_t stream) {
    (void)in_sizes; (void)n_in; (void)out_size; (void)ws_size;
    const int*   u_idx     = (const int*)d_in[0];
    const int*   r_idx     = (const int*)d_in[1];
    const int*   v_idx     = (const int*)d_in[2];
    const float* emb       = (const float*)d_in[3];
    const float* rel_diag  = (const float*)d_in[4];
    const float* rb1       = (const float*)d_in[5];
    const float* rb2       = (const float*)d_in[6];
    const float* bias_head = (const float*)d_in[7];
    const float* bias_tail = (const float*)d_in[8];
    const float* sigma     = (const float*)d_in[9];

    float*  ws_head = (float*)d_ws;                                  // B*64 f32
    float4* ws_scal = (float4*)((char*)d_ws + (size_t)Bq * DIM * sizeof(float));

    head_kernel<<<Bq, 32, 0, stream>>>(u_idx, r_idx, emb, rel_diag, rb1, rb2,
                                       bias_head, sigma, ws_head, ws_scal);
    score_kernel<<<Bq, 256, 0, stream>>>(v_idx, emb, bias_tail,
                                         ws_head, ws_scal, (float*)d_out);
}